// GCN_concat_90022514524461
// MI455X (gfx1250) — compile-verified
//
#include <hip/hip_runtime.h>
#include <math.h>

// ---------------------------------------------------------------------------
// Problem constants (from the reference)
// ---------------------------------------------------------------------------
#define N_NODES 50000
#define E_EDGES 800000
#define F1_DIM  128
#define F2_DIM  64
#define H_DIM   128
#define G_NUM   256
#define C_NUM   10

typedef float v2f __attribute__((ext_vector_type(2)));
typedef float v8f __attribute__((ext_vector_type(8)));

// ---------------------------------------------------------------------------
// WMMA f32 GEMM:  Out[N x 128] = A[N x K] @ W[K x 128]
// One wave per 16x16 tile; 8 waves per block -> 16 rows x 128 cols per block.
// Uses V_WMMA_F32_16X16X4_F32 (full fp32 precision; GEMM is not the
// bottleneck here so no need to downcast).
// ---------------------------------------------------------------------------
__global__ void __launch_bounds__(256)
gemm_wmma_f32(const float* __restrict__ A, const float* __restrict__ W,
              float* __restrict__ Out, int N, int K) {
  const int wave  = threadIdx.x >> 5;   // 0..7  -> column tile
  const int lane  = threadIdx.x & 31;
  const int m0    = blockIdx.x * 16;    // row tile (N is a multiple of 16)
  const int n0    = wave * 16;
  const int lhalf = lane >> 4;          // 0: lanes 0-15, 1: lanes 16-31
  const int lmod  = lane & 15;

  v8f c = {0.f, 0.f, 0.f, 0.f, 0.f, 0.f, 0.f, 0.f};

  const float* __restrict__ arow = A + (long)(m0 + lmod) * K;
  for (int k = 0; k < K; k += 4) {
    const int ka = k + 2 * lhalf;
    // A fragment: lanes 0-15 carry K={k,k+1}, lanes 16-31 carry K={k+2,k+3}
    v2f a;
    a.x = arow[ka];
    a.y = arow[ka + 1];
    // B fragment (mirror layout): col = n0+lmod
    v2f b;
    b.x = W[(long)(ka + 0) * H_DIM + n0 + lmod];
    b.y = W[(long)(ka + 1) * H_DIM + n0 + lmod];
    c = __builtin_amdgcn_wmma_f32_16x16x4_f32(
        /*neg_a=*/false, a, /*neg_b=*/false, b,
        /*c_mod=*/(short)0, c, /*reuse_a=*/false, /*reuse_b=*/false);
  }

  // C/D layout: VGPR r -> row m0 + r + 8*lhalf, col n0 + lmod
  float* __restrict__ ocol = Out + n0 + lmod;
#pragma unroll
  for (int r = 0; r < 8; ++r) {
    ocol[(long)(m0 + r + 8 * lhalf) * H_DIM] = c[r];
  }
}

// ---------------------------------------------------------------------------
// Degree / dinv
// ---------------------------------------------------------------------------
__global__ void fill_f32(float* __restrict__ p, float v, int n) {
  int i = blockIdx.x * blockDim.x + threadIdx.x;
  if (i < n) p[i] = v;
}

__global__ void deg_count(const int* __restrict__ dst, float* __restrict__ deg, int E) {
  int e = blockIdx.x * blockDim.x + threadIdx.x;
  if (e < E) atomicAdd(&deg[dst[e]], 1.0f);
}

__global__ void deg_to_rsqrt(float* __restrict__ deg, int N) {
  int i = blockIdx.x * blockDim.x + threadIdx.x;
  if (i < N) deg[i] = rsqrtf(deg[i]);   // deg >= 1 always
}

// ---------------------------------------------------------------------------
// agg[n,:] = hlin[n,:] * dinv[n]^2   (self-loop message; full overwrite)
// ---------------------------------------------------------------------------
__global__ void self_loop_init(const float* __restrict__ hlin,
                               const float* __restrict__ dinv,
                               float* __restrict__ agg, int N) {
  int i = blockIdx.x * blockDim.x + threadIdx.x;   // over N*128
  if (i >= N * H_DIM) return;
  int n = i >> 7;
  float di = dinv[n];
  agg[i] = hlin[i] * di * di;
}

// ---------------------------------------------------------------------------
// Edge aggregation: one wave per edge, lane owns a float4 slice of 128 feats.
// agg[dst,:] += hlin[src,:] * dinv[src]*dinv[dst]  via f32 atomics (L2).
// ---------------------------------------------------------------------------
__global__ void __launch_bounds__(256)
edge_aggregate(const float* __restrict__ hlin, const int* __restrict__ src,
               const int* __restrict__ dst, const float* __restrict__ dinv,
               float* __restrict__ agg, int E) {
  int e = blockIdx.x * (blockDim.x >> 5) + (threadIdx.x >> 5);
  if (e >= E) return;
  int lane = threadIdx.x & 31;
  int s = src[e], d = dst[e];
  float norm = dinv[s] * dinv[d];
  float4 v = ((const float4*)(hlin + (long)s * H_DIM))[lane];
  float* ad = agg + (long)d * H_DIM + lane * 4;
  atomicAdd(ad + 0, v.x * norm);
  atomicAdd(ad + 1, v.y * norm);
  atomicAdd(ad + 2, v.z * norm);
  atomicAdd(ad + 3, v.w * norm);
}

// ---------------------------------------------------------------------------
// out[n,f] = relu(agg[n,f] + bias[f])   (in-place safe)
// ---------------------------------------------------------------------------
__global__ void bias_relu(const float* __restrict__ agg, const float* __restrict__ bias,
                          float* __restrict__ out, int N) {
  int i = blockIdx.x * blockDim.x + threadIdx.x;
  if (i >= N * H_DIM) return;
  out[i] = fmaxf(agg[i] + bias[i & (H_DIM - 1)], 0.0f);
}

// ---------------------------------------------------------------------------
// Mean pool: atomic sums + counts, then divide.
// ---------------------------------------------------------------------------
__global__ void __launch_bounds__(256)
pool_sum(const float* __restrict__ h, const int* __restrict__ batch,
         float* __restrict__ psum, float* __restrict__ pcnt, int N) {
  int n = blockIdx.x * (blockDim.x >> 5) + (threadIdx.x >> 5);
  if (n >= N) return;
  int lane = threadIdx.x & 31;
  int g = batch[n];
  float4 v = ((const float4*)(h + (long)n * H_DIM))[lane];
  float* pg = psum + (long)g * H_DIM + lane * 4;
  atomicAdd(pg + 0, v.x);
  atomicAdd(pg + 1, v.y);
  atomicAdd(pg + 2, v.z);
  atomicAdd(pg + 3, v.w);
  if (lane == 0) atomicAdd(&pcnt[g], 1.0f);
}

__global__ void pool_div(const float* __restrict__ psum, const float* __restrict__ pcnt,
                         float* __restrict__ pmean) {
  int i = blockIdx.x * blockDim.x + threadIdx.x;   // over G*128
  if (i >= G_NUM * H_DIM) return;
  pmean[i] = psum[i] / fmaxf(pcnt[i >> 7], 1.0f);
}

// ---------------------------------------------------------------------------
// Final head: logits = [g1|g2] @ W3 + b3 ; log_softmax.  One wave per graph.
// ---------------------------------------------------------------------------
__global__ void __launch_bounds__(32)
final_head(const float* __restrict__ g1, const float* __restrict__ g2,
           const float* __restrict__ W3, const float* __restrict__ b3,
           float* __restrict__ out) {
  int g = blockIdx.x;
  int lane = threadIdx.x;
  float acc[C_NUM];
#pragma unroll
  for (int c = 0; c < C_NUM; ++c) acc[c] = 0.0f;

  for (int k = lane; k < 2 * H_DIM; k += 32) {
    float v = (k < H_DIM) ? g1[g * H_DIM + k] : g2[g * H_DIM + (k - H_DIM)];
    const float* wrow = W3 + k * C_NUM;
#pragma unroll
    for (int c = 0; c < C_NUM; ++c) acc[c] += v * wrow[c];
  }
#pragma unroll
  for (int c = 0; c < C_NUM; ++c) {
#pragma unroll
    for (int off = 16; off > 0; off >>= 1)
      acc[c] += __shfl_xor(acc[c], off, 32);
  }
  if (lane == 0) {
    float logits[C_NUM], mx = -INFINITY;
#pragma unroll
    for (int c = 0; c < C_NUM; ++c) {
      logits[c] = acc[c] + b3[c];
      mx = fmaxf(mx, logits[c]);
    }
    float se = 0.0f;
#pragma unroll
    for (int c = 0; c < C_NUM; ++c) se += expf(logits[c] - mx);
    float lse = mx + logf(se);
#pragma unroll
    for (int c = 0; c < C_NUM; ++c) out[g * C_NUM + c] = logits[c] - lse;
  }
}

// ---------------------------------------------------------------------------
// Host-side orchestration
// ---------------------------------------------------------------------------
static void run_side(const float* x, int F, const int* src, const int* dst,
                     const int* batch,
                     const float* conv_w, const float* conv_b,
                     const float* convs_w, const float* convs_b,
                     const float* lin_w, const float* lin_b,
                     float* dinv, float* B0, float* B1,
                     float* psum, float* pcnt, float* pmean, float* gtmp,
                     float* g_out, hipStream_t stream) {
  const int N = N_NODES, E = E_EDGES;
  const int nThreads = 256;
  const int nNF = N * H_DIM;                      // 6.4M
  const int gridNF = (nNF + nThreads - 1) / nThreads;

  // degrees -> dinv
  fill_f32<<<(N + 255) / 256, 256, 0, stream>>>(dinv, 1.0f, N);
  deg_count<<<(E + 255) / 256, 256, 0, stream>>>(dst, dinv, E);
  deg_to_rsqrt<<<(N + 255) / 256, 256, 0, stream>>>(dinv, N);

  const float* Xcur = x;
  int Kcur = F;
  for (int l = 0; l < 3; ++l) {
    const float* Wl = (l == 0) ? conv_w : (convs_w + (long)(l - 1) * H_DIM * H_DIM);
    const float* bl = (l == 0) ? conv_b : (convs_b + (long)(l - 1) * H_DIM);

    gemm_wmma_f32<<<N / 16, 256, 0, stream>>>(Xcur, Wl, B0, N, Kcur);
    self_loop_init<<<gridNF, nThreads, 0, stream>>>(B0, dinv, B1, N);
    edge_aggregate<<<(E + 7) / 8, 256, 0, stream>>>(B0, src, dst, dinv, B1, E);
    bias_relu<<<gridNF, nThreads, 0, stream>>>(B1, bl, B1, N);

    Xcur = B1;
    Kcur = H_DIM;
  }

  // mean pool
  fill_f32<<<(G_NUM * H_DIM + 255) / 256, 256, 0, stream>>>(psum, 0.0f, G_NUM * H_DIM);
  fill_f32<<<1, 256, 0, stream>>>(pcnt, 0.0f, G_NUM);
  pool_sum<<<(N + 7) / 8, 256, 0, stream>>>(B1, batch, psum, pcnt, N);
  pool_div<<<(G_NUM * H_DIM + 255) / 256, 256, 0, stream>>>(psum, pcnt, pmean);

  // g = relu(pmean @ lin_w + lin_b)   (G=256 rows -> 16 tiles)
  gemm_wmma_f32<<<G_NUM / 16, 256, 0, stream>>>(pmean, lin_w, gtmp, G_NUM, H_DIM);
  bias_relu<<<(G_NUM * H_DIM + 255) / 256, 256, 0, stream>>>(gtmp, lin_b, g_out, G_NUM);
}

extern "C" void kernel_launch(void* const* d_in, const int* in_sizes, int n_in,
                              void* d_out, int out_size, void* d_ws, size_t ws_size,
                              hipStream_t stream) {
  const float* x1       = (const float*)d_in[0];
  const int*   ei1      = (const int*)d_in[1];
  const int*   batch1   = (const int*)d_in[2];
  const float* x2       = (const float*)d_in[3];
  const int*   ei2      = (const int*)d_in[4];
  const int*   batch2   = (const int*)d_in[5];
  const float* conv1_w  = (const float*)d_in[6];
  const float* conv1_b  = (const float*)d_in[7];
  const float* convs1_w = (const float*)d_in[8];
  const float* convs1_b = (const float*)d_in[9];
  const float* conv2_w  = (const float*)d_in[10];
  const float* conv2_b  = (const float*)d_in[11];
  const float* convs2_w = (const float*)d_in[12];
  const float* convs2_b = (const float*)d_in[13];
  const float* lin1_w   = (const float*)d_in[14];
  const float* lin1_b   = (const float*)d_in[15];
  const float* lin2_w   = (const float*)d_in[16];
  const float* lin2_b   = (const float*)d_in[17];
  const float* lin3_w   = (const float*)d_in[18];
  const float* lin3_b   = (const float*)d_in[19];

  const int* src1 = ei1;
  const int* dst1 = ei1 + E_EDGES;
  const int* src2 = ei2;
  const int* dst2 = ei2 + E_EDGES;

  // workspace layout (floats); all offsets multiples of 16 bytes
  float* ws    = (float*)d_ws;
  float* dinv  = ws;                         // 50000 (padded to 50048)
  float* B0    = ws + 50048;                 // 6,400,000  (hlin)
  float* B1    = B0 + (long)N_NODES * H_DIM; // 6,400,000  (agg / features)
  float* psum  = B1 + (long)N_NODES * H_DIM; // 32768
  float* pcnt  = psum + G_NUM * H_DIM;       // 256
  float* pmean = pcnt + G_NUM;               // 32768
  float* gtmp  = pmean + G_NUM * H_DIM;      // 32768

  float* out   = (float*)d_out;
  float* g1out = out;                        // [G, H]
  float* g2out = out + G_NUM * H_DIM;        // [G, H]
  float* lsout = out + 2 * G_NUM * H_DIM;    // [G, C]

  run_side(x1, F1_DIM, src1, dst1, batch1, conv1_w, conv1_b, convs1_w, convs1_b,
           lin1_w, lin1_b, dinv, B0, B1, psum, pcnt, pmean, gtmp, g1out, stream);
  run_side(x2, F2_DIM, src2, dst2, batch2, conv2_w, conv2_b, convs2_w, convs2_b,
           lin2_w, lin2_b, dinv, B0, B1, psum, pcnt, pmean, gtmp, g2out, stream);

  final_head<<<G_NUM, 32, 0, stream>>>(g1out, g2out, lin3_w, lin3_b, lsout);
}